// ParallelTaylorDriver_49323404427867
// MI455X (gfx1250) — compile-verified
//
#include <hip/hip_runtime.h>
#include <hip/hip_bf16.h>
#include <math.h>
#include <stdint.h>

#define CHUNKS            16
#define RES_DIM           8192
#define BLOCKS_PER_CHUNK  64
#define THREADS           256

// ---------------------------------------------------------------------------
// Kernel 1: zero the global accumulator in workspace (must run every launch:
// the harness does not re-initialize d_ws between timed replays).
// ---------------------------------------------------------------------------
__global__ __launch_bounds__(THREADS)
void zero_acc_kernel(float* __restrict__ acc, int n) {
    int i = blockIdx.x * blockDim.x + threadIdx.x;
    if (i < n) acc[i] = 0.0f;
}

// ---------------------------------------------------------------------------
// Kernel 2: COO SpMV per chunk.
//   - Stage the chunk's 32KB state vector into LDS with CDNA5 async
//     global->LDS B128 copies (ASYNCcnt tracked, s_wait_asynccnt). RT hint:
//     the same 32KB is re-read by all 64 blocks of the chunk -> keep in L2.
//   - Stream the COO arrays (~258MB, read exactly once) with NON-TEMPORAL
//     loads so the one-shot stream doesn't thrash the 192MB L2.
//   - Random gather state[c] from LDS, v * state[c], ds_add_f32 scatter into
//     a 32KB LDS accumulator.
//   - Flush the LDS accumulator with non-returning global_atomic_add_f32.
// 64KB LDS per workgroup -> 5 concurrent workgroups per 320KB WGP.
// ---------------------------------------------------------------------------
__global__ __launch_bounds__(THREADS)
void coo_spmv_kernel(const float* __restrict__ wr_vals,
                     const int*   __restrict__ wr_rows,
                     const int*   __restrict__ wr_cols,
                     const float* __restrict__ res_state,
                     float*       __restrict__ acc,
                     int nse) {
    __shared__ __attribute__((aligned(16))) float sState[RES_DIM];
    __shared__ __attribute__((aligned(16))) float sAcc[RES_DIM];

    const int tid   = threadIdx.x;
    const int chunk = blockIdx.y;

    // --- Stage state[chunk] into LDS via async global->LDS copies ----------
    {
        const float* src = res_state + (size_t)chunk * RES_DIM;
        const unsigned long long base = (unsigned long long)(uintptr_t)src;
        // 256 threads * 16B = 4KB per sweep; 8 sweeps for 32KB.
        for (int i = tid * 4; i < RES_DIM; i += THREADS * 4) {
            // Generic LDS pointer truncates to the raw LDS byte address
            // (its low 32 bits).
            unsigned ldsAddr = (unsigned)(uintptr_t)(&sState[i]);
            unsigned byteOff = (unsigned)(i * 4);
            asm volatile("global_load_async_to_lds_b128 %0, %1, %2"
                         :: "v"(ldsAddr), "v"(byteOff), "s"(base)
                         : "memory");
        }
        // Zero the LDS accumulator while the async copies are in flight.
        for (int i = tid; i < RES_DIM; i += THREADS) sAcc[i] = 0.0f;
        asm volatile("s_wait_asynccnt 0x0" ::: "memory");
    }
    __syncthreads();

    // --- Stream this block's slice of the COO arrays (non-temporal) --------
    const size_t cbase = (size_t)chunk * (size_t)nse;
    const float* vals = wr_vals + cbase;
    const int*   rows = wr_rows + cbase;
    const int*   cols = wr_cols + cbase;

    const int per_block = (nse + BLOCKS_PER_CHUNK - 1) / BLOCKS_PER_CHUNK;
    const int start = blockIdx.x * per_block;
    const int end   = min(start + per_block, nse);

    for (int i = start + tid; i < end; i += THREADS) {
        float v = __builtin_nontemporal_load(&vals[i]);   // th:NT
        int   r = __builtin_nontemporal_load(&rows[i]);   // th:NT
        int   c = __builtin_nontemporal_load(&cols[i]);   // th:NT
        atomicAdd(&sAcc[r], v * sState[c]);   // ds_add_f32 (no return)
    }
    __syncthreads();

    // --- Flush LDS partial into the global accumulator ---------------------
    float* gacc = acc + (size_t)chunk * RES_DIM;
    for (int i = tid; i < RES_DIM; i += THREADS) {
        atomicAdd(&gacc[i], sAcc[i]);         // global_atomic_add_f32 (no rtn)
    }
}

// ---------------------------------------------------------------------------
// Kernel 3: out = Horner(tanh-Taylor coeffs, acc + proj)
// ---------------------------------------------------------------------------
__global__ __launch_bounds__(THREADS)
void finalize_kernel(const float* __restrict__ acc,
                     const float* __restrict__ proj,
                     float*       __restrict__ out,
                     int n) {
    int i = blockIdx.x * blockDim.x + threadIdx.x;
    if (i >= n) return;

    const float t  = tanhf(1.6f);
    const float t2 = t * t;
    const float t4 = t2 * t2;
    const float c0 = t;
    const float c1 = 1.0f - t2;
    const float c2 = t2 * t - t;
    const float c3 = -t4 + (4.0f / 3.0f) * t2 - (1.0f / 3.0f);
    const float c4 = (t / 3.0f) * (3.0f * t4 - 5.0f * t2 + 2.0f);
    const float c5 = -t4 * t2 + 2.0f * t4 - (17.0f / 15.0f) * t2 + (2.0f / 15.0f);

    float z = acc[i] + proj[i];
    float r = c5;
    r = fmaf(z, r, c4);
    r = fmaf(z, r, c3);
    r = fmaf(z, r, c2);
    r = fmaf(z, r, c1);
    r = fmaf(z, r, c0);
    out[i] = r;
}

// ---------------------------------------------------------------------------
extern "C" void kernel_launch(void* const* d_in, const int* in_sizes, int n_in,
                              void* d_out, int out_size, void* d_ws, size_t ws_size,
                              hipStream_t stream) {
    const float* proj  = (const float*)d_in[0];  // (CHUNKS, RES_DIM) f32
    const float* state = (const float*)d_in[1];  // (CHUNKS, RES_DIM) f32
    const float* vals  = (const float*)d_in[2];  // (CHUNKS, NSE)     f32
    const int*   rows  = (const int*)  d_in[3];  // (CHUNKS, NSE)     i32
    const int*   cols  = (const int*)  d_in[4];  // (CHUNKS, NSE)     i32
    float* out = (float*)d_out;                  // (CHUNKS, RES_DIM) f32
    float* acc = (float*)d_ws;                   // 512KB accumulator

    const int nse = in_sizes[2] / CHUNKS;
    const int n   = CHUNKS * RES_DIM;

    zero_acc_kernel<<<(n + THREADS - 1) / THREADS, THREADS, 0, stream>>>(acc, n);

    dim3 grid(BLOCKS_PER_CHUNK, CHUNKS);
    coo_spmv_kernel<<<grid, THREADS, 0, stream>>>(vals, rows, cols, state, acc, nse);

    finalize_kernel<<<(n + THREADS - 1) / THREADS, THREADS, 0, stream>>>(acc, proj, out, n);
}